// MC_81166291960419
// MI455X (gfx1250) — compile-verified
//
#include <hip/hip_runtime.h>
#include <cstdint>

#define MBK 8
#define PRNG 8
#define CROP 17
#define HDIM 512
#define WDIM 512
#define NBR 64
#define NBC 64
#define NFRM 16
#define NPAIR 14
#define OUTD 478
#define WIN 24            // MBK + 2*PRNG
#define WAVES_PER_WG 8
#define LARGE_SUM (65537.0f * 64.0f)   // reference LARGE, scaled: we compare sums not means

__device__ __forceinline__ float wave_reduce_sum(float s) {
  s += __shfl_xor(s, 16, 32);
  s += __shfl_xor(s, 8, 32);
  s += __shfl_xor(s, 4, 32);
  s += __shfl_xor(s, 2, 32);
  s += __shfl_xor(s, 1, 32);
  return s;
}

// One wave32 per 8x8 block: diamond-search motion + motion-compensated pred write.
__global__ __launch_bounds__(WAVES_PER_WG * 32)
void motion_comp_kernel(const float* __restrict__ x, float* __restrict__ out,
                        size_t predBase) {
  __shared__ float winbuf[WAVES_PER_WG][WIN * WIN];

  const int lane  = threadIdx.x & 31;
  const int wslot = threadIdx.x >> 5;
  const int wave  = blockIdx.x * WAVES_PER_WG + wslot;

  const int bx = wave & (NBC - 1);          // block col
  const int by = (wave >> 6) & (NBR - 1);   // block row
  const int bf = wave >> 12;                // (batch, pair)
  const int fk = bf % NPAIR;                // pair index 0..13
  const int b  = bf / NPAIR;                // batch

  const int i = by * MBK;
  const int j = bx * MBK;

  const float* imgI = x + ((size_t)(b * NFRM + fk) << 18);      // reference (t-1)
  const float* imgP = x + ((size_t)(b * NFRM + fk + 1) << 18);  // current   (t)

  float* w = winbuf[wslot];

  // ---- CDNA5 async global->LDS: stage 24x24 search window of imgI ----
  // 576 floats = 144 x 16B chunks; addresses 16B-aligned (j multiple of 8).
  // Border rows/cols clamped; clamped entries are only read by invalid
  // candidates whose cost is overridden with LARGE.
  #pragma unroll
  for (int it = 0; it < 5; ++it) {
    const int q = it * 32 + lane;
    if (q < (WIN * WIN) / 4) {
      const int row = q / 6;
      const int seg = q - row * 6;
      const int gr = min(max(i - PRNG + row, 0), HDIM - 1);
      const int gc = min(max(j - PRNG + seg * 4, 0), WDIM - 4);
      const float* src = imgI + ((size_t)gr << 9) + gc;
      const uint32_t ldsoff = (uint32_t)(uintptr_t)(w + row * WIN + seg * 4);
      asm volatile("global_load_async_to_lds_b128 %0, %1, off"
                   :: "v"(ldsoff), "v"((uint64_t)(uintptr_t)src)
                   : "memory");
    }
  }

  // ---- current 8x8 block: 2 px per lane (row = lane>>2, cols 2*(lane&3)) ----
  const int r = lane >> 2;
  const int c = (lane & 3) << 1;
  const float* curp = imgP + (size_t)(i + r) * WDIM + (j + c);
  const float c0a = curp[0];
  const float c0b = curp[1];

  // ASYNCcnt is per-wave: no workgroup barrier needed
  asm volatile("s_wait_asynccnt 0" ::: "memory");

  // SAD (sum of |diff| over 64 px) at absolute candidate top-left (cy,cx).
  // Fixed reduction order -> bit-identical results for identical positions,
  // which makes center-cost caching exact.
  auto sad = [&](int cy, int cx) -> float {
    const bool valid = (cy >= 0) & (cy <= HDIM - MBK) &
                       (cx >= 0) & (cx <= WDIM - MBK) &
                       (cy - i <= PRNG) & (i - cy <= PRNG) &
                       (cx - j <= PRNG) & (j - cx <= PRNG);
    const int wy = min(max(cy - (i - PRNG), 0), 2 * PRNG);
    const int wx = min(max(cx - (j - PRNG), 0), 2 * PRNG);
    const float a0 = w[(wy + r) * WIN + wx + c];
    const float a1 = w[(wy + r) * WIN + wx + c + 1];
    float s = fabsf(c0a - a0) + fabsf(c0b - a1);
    s = wave_reduce_sum(s);          // all lanes hold the full sum
    return valid ? s : LARGE_SUM;
  };

  // LDSP / SDSP (dx, dy) as in the reference; argmin keeps FIRST minimum.
  const int LX[9] = { 0, -1, 1, -2, 0, 2, -1, 1, 0 };
  const int LY[9] = {-2, -1,-1,  0, 0, 0,  1, 1, 2 };
  const int SX[5] = { 0, -1, 0, 1, 0 };
  const int SY[5] = {-1,  0, 0, 0, 1 };

  int y = i, xx = j;
  float centerCost = sad(i, j);                 // cost at current center (y,xx)
  int pt = (centerCost == 0.0f) ? 4 : -1;

  for (int step = 0; step < 16 && pt != 4; ++step) {
    float best = 3.0e38f;
    int bsel = 0;
    #pragma unroll
    for (int t = 0; t < 9; ++t) {
      // LDSP index 4 is (0,0): its cost is the cached center cost.
      const float ct = (t == 4) ? centerCost : sad(y + LY[t], xx + LX[t]);
      if (ct < best) { best = ct; bsel = t; }
    }
    y += LY[bsel];
    xx += LX[bsel];
    pt = bsel;
    centerCost = best;               // cost at the new center, bit-exact
  }
  {
    float best = 3.0e38f;
    int bsel = 0;
    #pragma unroll
    for (int t = 0; t < 5; ++t) {
      // SDSP index 2 is (0,0): cached center cost.
      const float ct = (t == 2) ? centerCost : sad(y + SY[t], xx + SX[t]);
      if (ct < best) { best = ct; bsel = t; }
    }
    y += SY[bsel];
    xx += SX[bsel];
  }

  const int dy = y - i;
  const int dx = xx - j;

  // ---- motion compensation (dynamic_slice clamp semantics) + crop write ----
  const int sy = min(max(i + dy, 0), HDIM - MBK);
  const int sx = min(max(j + dx, 0), WDIM - MBK);
  const float* sp = imgP + (size_t)(sy + r) * WDIM + (sx + c);
  const float o0 = sp[0];
  const float o1 = sp[1];

  const int R = i + r;
  const int C = j + c;
  if (R >= CROP && R < HDIM - CROP) {
    float* prow = out + predBase +
                  ((size_t)(b * NPAIR + fk) * OUTD + (size_t)(R - CROP)) * OUTD - CROP;
    if (C >= CROP && C < WDIM - CROP)         prow[C]     = o0;
    if (C + 1 >= CROP && C + 1 < WDIM - CROP) prow[C + 1] = o1;
  }
}

// target[b, k] = vid[b, k+2][17:495, 17:495]
__global__ __launch_bounds__(256)
void target_copy_kernel(const float* __restrict__ x, float* __restrict__ out, int n) {
  const int idx = blockIdx.x * blockDim.x + threadIdx.x;
  if (idx >= n) return;
  int t = idx;
  const int cc = t % OUTD; t /= OUTD;
  const int rr = t % OUTD; t /= OUTD;
  const int fk = t % NPAIR;
  const int b  = t / NPAIR;
  out[idx] = x[((size_t)(b * NFRM + fk + 2) << 18) +
               (size_t)(rr + CROP) * WDIM + (cc + CROP)];
}

extern "C" void kernel_launch(void* const* d_in, const int* in_sizes, int n_in,
                              void* d_out, int out_size, void* d_ws, size_t ws_size,
                              hipStream_t stream) {
  (void)n_in; (void)d_ws; (void)ws_size;
  const float* x = (const float*)d_in[0];
  float* out = (float*)d_out;

  const int B = in_sizes[0] / (NFRM * HDIM * WDIM);   // 4
  const size_t half = (size_t)out_size / 2;           // target size == pred size

  // targets: strided crop copy (pure bandwidth, trivially small vs 23.3 TB/s)
  {
    const int n = (int)half;
    const int grid = (n + 255) / 256;
    target_copy_kernel<<<grid, 256, 0, stream>>>(x, out, n);
  }
  // motion search + compensated pred
  {
    const int numWaves = B * NPAIR * NBR * NBC;       // 229376
    const int wgs = numWaves / WAVES_PER_WG;
    motion_comp_kernel<<<wgs, WAVES_PER_WG * 32, 0, stream>>>(x, out, half);
  }
}